// DistillPtCloud_37331855737595
// MI455X (gfx1250) — compile-verified
//
#include <hip/hip_runtime.h>

// Problem constants (from reference)
#define BB 16
#define HH 352
#define WW 1216
#define NN (HH * WW)          // 428032
#define PP 100000
#define TYPEIND_ 5
#define MAXD 40.0f

#define ELB 1024              // elements per block in count/compact (256 thr x 4)
#define NB  (NN / ELB)        // 418 blocks per batch (exact: 418*1024 = 428032)
#define TB  (BB * NB)         // 6688 total blocks

typedef __attribute__((ext_vector_type(2))) float v2f;
typedef __attribute__((ext_vector_type(8))) float v8f;

// ---------------------------------------------------------------------------
// Kernel 1: per-block count of selected elements (in permuted order)
// ---------------------------------------------------------------------------
__global__ void k_count(const int* __restrict__ sem, const float* __restrict__ dep,
                        const int* __restrict__ perm, int* __restrict__ bcnt) {
    int b = blockIdx.y, blk = blockIdx.x, t = threadIdx.x;
    int j0 = blk * ELB + t * 4;
    int base = b * NN;
    int cnt = 0;
#pragma unroll
    for (int e = 0; e < 4; ++e) {
        int q = perm[j0 + e];
        cnt += (sem[base + q] == TYPEIND_ && dep[base + q] < MAXD) ? 1 : 0;
    }
    __shared__ int sc[256];
    sc[t] = cnt;
    __syncthreads();
    for (int o = 128; o > 0; o >>= 1) {
        if (t < o) sc[t] += sc[t + o];
        __syncthreads();
    }
    if (t == 0) bcnt[b * NB + blk] = sc[0];
}

// ---------------------------------------------------------------------------
// Kernel 2: single-block exclusive scan of block counts; batch counts/offsets;
//           valid_batch_indicator written to output tail.
// ---------------------------------------------------------------------------
__global__ void k_scan(const int* __restrict__ bcnt, int* __restrict__ boff,
                       int* __restrict__ vni, int* __restrict__ bofB,
                       int* __restrict__ cmp, float* __restrict__ outInd) {
    __shared__ int sums[256];
    __shared__ int sTot;
    int t = threadIdx.x;
    const int chunk = (TB + 255) / 256;         // 27
    int beg = t * chunk;
    int end = beg + chunk; if (end > TB) end = TB;
    int s = 0;
    for (int i = beg; i < end; ++i) s += bcnt[i];
    sums[t] = s;
    __syncthreads();
    if (t == 0) {
        int run = 0;
        for (int i = 0; i < 256; ++i) { int v = sums[i]; sums[i] = run; run += v; }
        sTot = run;
    }
    __syncthreads();
    int run = sums[t];
    for (int i = beg; i < end; ++i) { int v = bcnt[i]; boff[i] = run; run += v; }
    __syncthreads();
    if (t < BB) {
        int st = boff[t * NB];
        int en = (t == BB - 1) ? sTot : boff[(t + 1) * NB];
        vni[t]  = en - st;
        bofB[t] = st;
        outInd[t] = (en > st) ? 1.0f : 0.0f;
    }
    if (t == 0 && sTot == 0) cmp[0] = 0;        // only slot readable when nothing valid
}

// ---------------------------------------------------------------------------
// Kernel 3: order-preserving compaction of linear indices (pi + b*N)
// ---------------------------------------------------------------------------
__global__ void k_compact(const int* __restrict__ sem, const float* __restrict__ dep,
                          const int* __restrict__ perm, const int* __restrict__ boff,
                          int* __restrict__ cmp) {
    int b = blockIdx.y, blk = blockIdx.x, t = threadIdx.x;
    int j0 = blk * ELB + t * 4;
    int base = b * NN;
    int pi[4]; int sl[4]; int cnt = 0;
#pragma unroll
    for (int e = 0; e < 4; ++e) {
        int q = perm[j0 + e];
        pi[e] = q;
        int ok = (sem[base + q] == TYPEIND_ && dep[base + q] < MAXD) ? 1 : 0;
        sl[e] = ok; cnt += ok;
    }
    __shared__ int sc[256];
    sc[t] = cnt;
    __syncthreads();
    // Hillis-Steele inclusive scan of per-thread counts
    for (int o = 1; o < 256; o <<= 1) {
        int v = (t >= o) ? sc[t - o] : 0;
        __syncthreads();
        sc[t] += v;
        __syncthreads();
    }
    int slot = boff[b * NB + blk] + sc[t] - cnt;
#pragma unroll
    for (int e = 0; e < 4; ++e) {
        if (sl[e]) cmp[slot++] = pi[e] + base;
    }
}

// ---------------------------------------------------------------------------
// Kernel 4: modulo sampling + gather + WMMA 16x16x4 FP32 camera transform.
// Each wave32 handles 16 points: lanes L and L+16 carry the same point's
// homogeneous data split across the B-matrix K-halves (ISA 16x4/4x16 layout).
// A = invcamK[b'] (wave-uniform), rows 0..3 used, rows 4..15 zero.
// D rows 0..2 on lanes 0..15 are the x/y/z outputs.
// ---------------------------------------------------------------------------
__global__ void k_sample(const float* __restrict__ bind, const int* __restrict__ vni,
                         const int* __restrict__ bofB, const int* __restrict__ cmp,
                         const float* __restrict__ dep, const float* __restrict__ K,
                         float* __restrict__ out) {
    int b    = blockIdx.y;
    int t    = threadIdx.x;
    int lane = t & 31;
    int wv   = t >> 5;
    int pRaw = blockIdx.x * 128 + wv * 16 + (lane & 15);
    bool pv  = pRaw < PP;
    int  p   = pv ? pRaw : (PP - 1);            // clamp so EXEC stays full for WMMA

    int   vn  = vni[b];
    float mmd = (vn == 0) ? 1.0f : (float)vn;
    float s   = fmodf(bind[b * PP + p], mmd);   // jnp.remainder, both operands >= 0
    if (vn != 0) s += (float)bofB[b];           // offsets * valid_batch_indicator
    int idx = (int)s;                           // exact: value < 2^24
    int pos = cmp[idx];

    int bp = pos / NN;
    int n  = pos - bp * NN;
    int y  = n / WW;
    int x  = n - y * WW;
    float d = dep[pos];                         // predDepth flat index == pos

    // A-matrix (16x4 f32): lane m = lane&15; K-halves split at lane 16
    const float* Kb = K + bp * 16;
    int mr = lane & 15;
    int k0 = (lane >> 4) << 1;                  // 0 or 2
    v2f av;
    if (mr < 4) { av.x = Kb[mr * 4 + k0]; av.y = Kb[mr * 4 + k0 + 1]; }
    else        { av.x = 0.0f;            av.y = 0.0f; }

    // B-matrix (4x16 f32): rows {0,2} in v0, {1,3} in v1, split at lane 16
    v2f bv;
    if (lane < 16) { bv.x = (float)x * d; bv.y = (float)y * d; }  // homo rows 0,1
    else           { bv.x = d;            bv.y = 1.0f;        }   // homo rows 2,3

    v8f cz = {};
    v8f dd = __builtin_amdgcn_wmma_f32_16x16x4_f32(
        /*neg_a=*/false, av, /*neg_b=*/false, bv,
        /*c_mod=*/(short)0, cz, /*reuse_a=*/false, /*reuse_b=*/false);

    if (lane < 16 && pv) {
        out[b * 3 * PP + 0 * PP + p] = dd[0];
        out[b * 3 * PP + 1 * PP + p] = dd[1];
        out[b * 3 * PP + 2 * PP + p] = dd[2];
    }
}

// ---------------------------------------------------------------------------
extern "C" void kernel_launch(void* const* d_in, const int* in_sizes, int n_in,
                              void* d_out, int out_size, void* d_ws, size_t ws_size,
                              hipStream_t stream) {
    const float* predDepth = (const float*)d_in[0];   // [B,1,H,W]
    const float* invcamK   = (const float*)d_in[1];   // [B,4,4]
    const int*   sem       = (const int*)  d_in[2];   // [B,1,H,W]
    const float* bind      = (const float*)d_in[3];   // [B,P]
    const int*   perm      = (const int*)  d_in[4];   // [N]
    float* out = (float*)d_out;                       // [B,3,P] ++ [B]

    // Workspace layout (ints): compact[B*N] | bcnt[TB] | boff[TB] | vni[B] | bofB[B]
    int* cmp  = (int*)d_ws;
    int* bcnt = cmp + (size_t)BB * NN;
    int* boff = bcnt + TB;
    int* vni  = boff + TB;
    int* bofB = vni + BB;

    k_count  <<<dim3(NB, BB), 256, 0, stream>>>(sem, predDepth, perm, bcnt);
    k_scan   <<<1, 256, 0, stream>>>(bcnt, boff, vni, bofB, cmp, out + 3 * BB * PP);
    k_compact<<<dim3(NB, BB), 256, 0, stream>>>(sem, predDepth, perm, boff, cmp);
    k_sample <<<dim3((PP + 127) / 128, BB), 256, 0, stream>>>(
        bind, vni, bofB, cmp, predDepth, invcamK, out);
}